// Grouped_multi_axis_Hadamard_Product_Attention_15401752723618
// MI455X (gfx1250) — compile-verified
//
#include <hip/hip_runtime.h>
#include <hip/hip_bf16.h>

typedef __attribute__((ext_vector_type(2))) float v2f;
typedef __attribute__((ext_vector_type(8))) float v8f;

#define HW_SHIFT 16  // H*W = 256*256 = 65536

__device__ __forceinline__ float gelu_f(float v) {
  // exact (erf) GELU, matches jax.nn.gelu(approximate=False)
  return 0.5f * v * (1.0f + erff(v * 0.70710678118654752440f));
}

// bilinear sample (align_corners=True) from an 8x8 grid at integer pixel (ty,tx) in [0,255]
__device__ __forceinline__ float bisample8x8(const float* __restrict__ Pc, int ty, int tx) {
  const float s = 7.0f / 255.0f;
  float fy = ty * s, fx = tx * s;
  int y0 = (int)fy; if (y0 > 6) y0 = 6;
  int x0 = (int)fx; if (x0 > 6) x0 = 6;
  float wy = fy - (float)y0, wx = fx - (float)x0;
  float p00 = Pc[y0 * 8 + x0],       p01 = Pc[y0 * 8 + x0 + 1];
  float p10 = Pc[(y0 + 1) * 8 + x0], p11 = Pc[(y0 + 1) * 8 + x0 + 1];
  float a = p00 + (p01 - p00) * wx;
  float b = p10 + (p11 - p10) * wx;
  return a + (b - a) * wy;
}

// 1D align_corners sample from 8-vector at integer position t in [0,255]
__device__ __forceinline__ float sample8(const float* __restrict__ Pc, int t) {
  const float s = 7.0f / 255.0f;
  float f = t * s;
  int i0 = (int)f; if (i0 > 6) i0 = 6;
  float w = f - (float)i0;
  return Pc[i0] + (Pc[i0 + 1] - Pc[i0]) * w;
}

// ---------------------------------------------------------------------------
// att_xy[32][256][256]: bilinear-upsample params_xy, dw3x3+b, GELU, pw32x32+b
// one wave per pixel: lane == channel, matvec via __shfl
// ---------------------------------------------------------------------------
__global__ __launch_bounds__(256) void k_att_xy(
    const float* __restrict__ params_xy, const float* __restrict__ dw_w,
    const float* __restrict__ dw_b, const float* __restrict__ pw_w,
    const float* __restrict__ pw_b, float* __restrict__ att_xy) {
  int c   = threadIdx.x & 31;
  int pix = blockIdx.x * 8 + (threadIdx.x >> 5);  // 8192 blocks * 8 pixels
  int h = pix >> 8, w = pix & 255;
  const float* Pc = params_xy + c * 64;
  float acc = dw_b[c];
#pragma unroll
  for (int ky = 0; ky < 3; ++ky) {
    int hh = h + ky - 1;
    bool hok = (hh >= 0) && (hh < 256);
#pragma unroll
    for (int kx = 0; kx < 3; ++kx) {
      int ww = w + kx - 1;
      float v = (hok && ww >= 0 && ww < 256) ? bisample8x8(Pc, hh, ww) : 0.0f;
      acc += dw_w[(c * 3 + ky) * 3 + kx] * v;
    }
  }
  float g = gelu_f(acc);
  float out = pw_b[c];
#pragma unroll
  for (int cc = 0; cc < 32; ++cc) {
    float gv = __shfl(g, cc, 32);
    out += pw_w[c * 32 + cc] * gv;
  }
  att_xy[(c << HW_SHIFT) + pix] = out;  // out-channel == lane
}

// ---------------------------------------------------------------------------
// att_z[32][256] for the zx / zy branches (1D version of the above)
// ---------------------------------------------------------------------------
__global__ __launch_bounds__(256) void k_att_z(
    const float* __restrict__ paramz, const float* __restrict__ dw_w,
    const float* __restrict__ dw_b, const float* __restrict__ pw_w,
    const float* __restrict__ pw_b, float* __restrict__ att) {
  int c   = threadIdx.x & 31;
  int grp = threadIdx.x >> 5;  // 8 groups x 32 t-values
  const float* Pc = paramz + c * 8;
  for (int i = 0; i < 32; ++i) {
    int t = grp * 32 + i;
    float acc = dw_b[c];
#pragma unroll
    for (int k = 0; k < 3; ++k) {
      int u = t + k - 1;
      float v = (u >= 0 && u < 256) ? sample8(Pc, u) : 0.0f;
      acc += dw_w[c * 3 + k] * v;
    }
    float tt = gelu_f(acc);
    float out = pw_b[c];
#pragma unroll
    for (int cc = 0; cc < 32; ++cc) {
      float gv = __shfl(tt, cc, 32);
      out += pw_w[c * 32 + cc] * gv;
    }
    att[c * 256 + t] = out;
  }
}

// ---------------------------------------------------------------------------
// Stage 1: LN1 over channels; store n1[128]; t4 = gelu(pw4(n1[96:128]))
// ---------------------------------------------------------------------------
__global__ __launch_bounds__(256) void k_s1(
    const float* __restrict__ x, const float* __restrict__ n1w,
    const float* __restrict__ n1b, const float* __restrict__ pw4_w,
    const float* __restrict__ pw4_b, float* __restrict__ n1,
    float* __restrict__ t4) {
  int p = blockIdx.x * 256 + threadIdx.x;  // 262144 pixels, lanes adjacent in w
  int b = p >> 16;
  int hw = p & 65535;
  const float* xb = x + ((size_t)(b * 128) << HW_SHIFT) + hw;
  float sum = 0.f, ss = 0.f;
#pragma unroll
  for (int c = 0; c < 128; ++c) {
    float v = xb[(size_t)c << HW_SHIFT];
    sum += v; ss += v * v;
  }
  float mean = sum * (1.0f / 128.0f);
  float var  = ss * (1.0f / 128.0f) - mean * mean;
  float inv  = rsqrtf(var + 1e-6f);
  float n4[32];
  float* n1p = n1 + ((size_t)(b * 128) << HW_SHIFT) + hw;
#pragma unroll
  for (int c = 0; c < 128; ++c) {
    float v = (xb[(size_t)c << HW_SHIFT] - mean) * inv * n1w[c] + n1b[c];
    n1p[(size_t)c << HW_SHIFT] = v;
    if (c >= 96) n4[c - 96] = v;
  }
  float* t4p = t4 + ((size_t)(b * 32) << HW_SHIFT) + hw;
#pragma unroll
  for (int o = 0; o < 32; ++o) {
    float acc = pw4_b[o];
#pragma unroll
    for (int c = 0; c < 32; ++c) acc += pw4_w[o * 32 + c] * n4[c];
    t4p[(size_t)o << HW_SHIFT] = gelu_f(acc);
  }
}

// ---------------------------------------------------------------------------
// Stage 2: Hadamard groups + dw3x3(t4) + LN2 + residual; z overwrites n1
// (each thread touches only its own pixel in n1 -> in-place is race-free)
// ---------------------------------------------------------------------------
__global__ __launch_bounds__(256) void k_s2(
    float* __restrict__ n1z, const float* __restrict__ t4,
    const float* __restrict__ axy, const float* __restrict__ azx,
    const float* __restrict__ azy, const float* __restrict__ dww,
    const float* __restrict__ dwb, const float* __restrict__ n2w,
    const float* __restrict__ n2b) {
  int p = blockIdx.x * 256 + threadIdx.x;
  int b = p >> 16;
  int hw = p & 65535;
  int h = hw >> 8, w = hw & 255;
  float* n1p = n1z + ((size_t)(b * 128) << HW_SHIFT) + hw;
  float cc[128];
#pragma unroll
  for (int c = 0; c < 32; ++c)
    cc[c] = n1p[(size_t)c << HW_SHIFT] * axy[(c << HW_SHIFT) + hw];
#pragma unroll
  for (int c = 0; c < 32; ++c)
    cc[32 + c] = n1p[(size_t)(32 + c) << HW_SHIFT] * azx[c * 256 + h];
#pragma unroll
  for (int c = 0; c < 32; ++c)
    cc[64 + c] = n1p[(size_t)(64 + c) << HW_SHIFT] * azy[c * 256 + w];
  const float* t4b = t4 + ((size_t)(b * 32) << HW_SHIFT);
#pragma unroll
  for (int o = 0; o < 32; ++o) {
    float acc = dwb[o];
#pragma unroll
    for (int ky = 0; ky < 3; ++ky) {
      int hh = h + ky - 1;
      bool hok = (hh >= 0) && (hh < 256);
#pragma unroll
      for (int kx = 0; kx < 3; ++kx) {
        int ww = w + kx - 1;
        float v = (hok && ww >= 0 && ww < 256)
                      ? t4b[((size_t)o << HW_SHIFT) + (hh << 8) + ww]
                      : 0.0f;
        acc += dww[(o * 3 + ky) * 3 + kx] * v;
      }
    }
    cc[96 + o] = acc;
  }
  float sum = 0.f, ss = 0.f;
#pragma unroll
  for (int c = 0; c < 128; ++c) { sum += cc[c]; ss += cc[c] * cc[c]; }
  float mean = sum * (1.0f / 128.0f);
  float var  = ss * (1.0f / 128.0f) - mean * mean;
  float inv  = rsqrtf(var + 1e-6f);
#pragma unroll
  for (int c = 0; c < 128; ++c) {
    float res = n1p[(size_t)c << HW_SHIFT];
    n1p[(size_t)c << HW_SHIFT] = (cc[c] - mean) * inv * n2w[c] + n2b[c] + res;
  }
}

// ---------------------------------------------------------------------------
// Stage 3: d = gelu(dw3x3(z)) into LDS, then 128x128 projection via
// V_WMMA_F32_16X16X4_F32. 128-pixel row tile, 8 waves x 16 pixels each.
// ---------------------------------------------------------------------------
__global__ __launch_bounds__(256) void k_s3(
    const float* __restrict__ z, const float* __restrict__ dww,
    const float* __restrict__ dwb, const float* __restrict__ pww,
    const float* __restrict__ pwb, float* __restrict__ out) {
  extern __shared__ float lds[];
  float* dT = lds;              // [128 px][130]  (pad->2-bank stride)
  float* Wt = lds + 128 * 130;  // [128 c ][130]  Wt[c][o] = W[o][c]
  int blk   = blockIdx.x;       // 2048 = B(4) * H(256) * (W/128 = 2)
  int wbase = (blk & 1) << 7;
  int h     = (blk >> 1) & 255;
  int b     = blk >> 9;
  int tid   = threadIdx.x;

  for (int i = tid; i < 128 * 128; i += 256) {  // coalesced read, padded write
    int o = i >> 7, c = i & 127;
    Wt[c * 130 + o] = pww[i];
  }

  const float* zb = z + ((size_t)(b * 128) << HW_SHIFT);
  int px = tid & 127;           // fixed pixel per thread -> coalesced z reads
  int wg = wbase + px;
  for (int it = 0; it < 64; ++it) {
    int ch = it * 2 + (tid >> 7);
    const float* zc = zb + ((size_t)ch << HW_SHIFT);
    float acc = dwb[ch];
#pragma unroll
    for (int ky = 0; ky < 3; ++ky) {
      int hh = h + ky - 1;
      bool hok = (hh >= 0) && (hh < 256);
#pragma unroll
      for (int kx = 0; kx < 3; ++kx) {
        int ww = wg + kx - 1;
        float v = (hok && ww >= 0 && ww < 256) ? zc[(hh << 8) + ww] : 0.0f;
        acc += dww[ch * 9 + ky * 3 + kx] * v;
      }
    }
    dT[px * 130 + ch] = gelu_f(acc);
  }
  __syncthreads();

  int wv = tid >> 5, lane = tid & 31;
  int half = lane >> 4, l15 = lane & 15;
  v8f acc[8] = {};  // 8 N-tiles of 16 -> all 128 output channels
  const float* arow = &dT[(wv * 16 + l15) * 130];  // this wave's pixel rows
  for (int kk = 0; kk < 128; kk += 4) {
    int k0 = kk + half * 2;  // A layout: lanes 0-15 K={kk,kk+1}, 16-31 K={kk+2,kk+3}
    v2f a = *(const v2f*)(arow + k0);
#pragma unroll
    for (int nt = 0; nt < 8; ++nt) {
      int n = nt * 16 + l15;
      v2f bb;
      bb.x = Wt[k0 * 130 + n];
      bb.y = Wt[(k0 + 1) * 130 + n];
      acc[nt] = __builtin_amdgcn_wmma_f32_16x16x4_f32(
          false, a, false, bb, (short)0, acc[nt], false, false);
    }
  }

  // stage results through this wave's own dT rows (exclusive -> no barrier),
  // so global stores are contiguous in w instead of 4B channel-scatter.
#pragma unroll
  for (int nt = 0; nt < 8; ++nt) {
    int o = nt * 16 + l15;
    float bias = pwb[o];
#pragma unroll
    for (int r = 0; r < 8; ++r) {
      int opx = wv * 16 + half * 8 + r;  // C/D layout: lanes16-31 hold M=r+8
      dT[opx * 130 + o] = acc[nt][r] + bias;
    }
  }
  float* ob = out + ((size_t)(b * 128) << HW_SHIFT) + (h << 8) + wbase;
  int spx = wv * 16 + l15;
  for (int j = 0; j < 64; ++j) {
    int o = j * 2 + half;
    ob[((size_t)o << HW_SHIFT) + spx] = dT[spx * 130 + o];
  }
}

// ---------------------------------------------------------------------------
extern "C" void kernel_launch(void* const* d_in, const int* in_sizes, int n_in,
                              void* d_out, int out_size, void* d_ws, size_t ws_size,
                              hipStream_t stream) {
  (void)in_sizes; (void)n_in; (void)out_size; (void)ws_size;
  const float* x         = (const float*)d_in[0];
  const float* norm1_w   = (const float*)d_in[1];
  const float* norm1_b   = (const float*)d_in[2];
  const float* params_xy = (const float*)d_in[3];
  const float* cxy_dw_w  = (const float*)d_in[4];
  const float* cxy_dw_b  = (const float*)d_in[5];
  const float* cxy_pw_w  = (const float*)d_in[6];
  const float* cxy_pw_b  = (const float*)d_in[7];
  const float* params_zx = (const float*)d_in[8];
  const float* czx_dw_w  = (const float*)d_in[9];
  const float* czx_dw_b  = (const float*)d_in[10];
  const float* czx_pw_w  = (const float*)d_in[11];
  const float* czx_pw_b  = (const float*)d_in[12];
  const float* params_zy = (const float*)d_in[13];
  const float* czy_dw_w  = (const float*)d_in[14];
  const float* czy_dw_b  = (const float*)d_in[15];
  const float* czy_pw_w  = (const float*)d_in[16];
  const float* czy_pw_b  = (const float*)d_in[17];
  const float* dw_pw_w   = (const float*)d_in[18];
  const float* dw_pw_b   = (const float*)d_in[19];
  const float* dw_dw_w   = (const float*)d_in[20];
  const float* dw_dw_b   = (const float*)d_in[21];
  const float* norm2_w   = (const float*)d_in[22];
  const float* norm2_b   = (const float*)d_in[23];
  const float* ldw_dw_w  = (const float*)d_in[24];
  const float* ldw_dw_b  = (const float*)d_in[25];
  const float* ldw_pw_w  = (const float*)d_in[26];
  const float* ldw_pw_b  = (const float*)d_in[27];

  float* ws  = (float*)d_ws;
  float* n1  = ws;                                  // 4*128*65536 (reused as z in-place)
  float* t4  = n1 + (size_t)4 * 128 * 65536;        // 4*32*65536
  float* axy = t4 + (size_t)4 * 32 * 65536;         // 32*65536
  float* azx = axy + (size_t)32 * 65536;            // 32*256
  float* azy = azx + 32 * 256;                      // 32*256

  k_att_xy<<<8192, 256, 0, stream>>>(params_xy, cxy_dw_w, cxy_dw_b, cxy_pw_w,
                                     cxy_pw_b, axy);
  k_att_z<<<1, 256, 0, stream>>>(params_zx, czx_dw_w, czx_dw_b, czx_pw_w,
                                 czx_pw_b, azx);
  k_att_z<<<1, 256, 0, stream>>>(params_zy, czy_dw_w, czy_dw_b, czy_pw_w,
                                 czy_pw_b, azy);
  k_s1<<<1024, 256, 0, stream>>>(x, norm1_w, norm1_b, dw_pw_w, dw_pw_b, n1, t4);
  k_s2<<<1024, 256, 0, stream>>>(n1, t4, axy, azx, azy, dw_dw_w, dw_dw_b,
                                 norm2_w, norm2_b);
  k_s3<<<2048, 256, 2 * 128 * 130 * sizeof(float), stream>>>(
      n1, ldw_dw_w, ldw_dw_b, ldw_pw_w, ldw_pw_b, (float*)d_out);
}